// GlobalNormFlipFlopCatMod_7662221656151
// MI455X (gfx1250) — compile-verified
//
#include <hip/hip_runtime.h>
#include <math.h>

// Problem constants (from the reference)
#define TT      2000
#define NB      128
#define INSIZE  256
#define SIZE_Y  43     // linear output channels
#define NPAD    48     // padded to 3 WMMA N-tiles
#define OUTC    46     // output channels: 40 trans + 2 + 2 + 1 + 1
#define NROWS   (TT * NB)

// Parallel-scan decomposition of the T=2000 recurrence
#define SEG     50     // number of segments
#define SEGLEN  40     // steps per segment (SEG*SEGLEN == TT)

typedef __attribute__((ext_vector_type(2))) float v2f;
typedef __attribute__((ext_vector_type(8))) float v8f;

// ---------------------------------------------------------------------------
// Kernel 1: y = x @ W^T + b via V_WMMA_F32_16X16X4_F32.
//   - writes raw y[:, 0:40] into out cols 0..39 (logZ correction applied later)
//   - writes final log-softmax mod outputs into out cols 40..45
// Block: 256 threads = 8 waves; each wave computes a 16-row x 48-col tile.
// ---------------------------------------------------------------------------
__global__ __launch_bounds__(256) void gemm_wmma_kernel(
    const float* __restrict__ x, const float* __restrict__ W,
    const float* __restrict__ b, float* __restrict__ out)
{
    // W^T staged as K-pairs: sW[(k>>1)*98 + n*2 + (k&1)], n padded to 49 float2
    __shared__ __align__(16) float sW[128 * 98];   // ~49 KB
    __shared__ float sB[NPAD];
    __shared__ float sMod[8][16][4];               // per-wave y[.,40..42]

    const int tid = threadIdx.x;

    for (int e = tid; e < NPAD * INSIZE; e += 256) {
        int n = e >> 8;          // output channel
        int k = e & 255;         // reduction index (coalesced across threads)
        float v = (n < SIZE_Y) ? W[n * INSIZE + k] : 0.0f;
        sW[(k >> 1) * 98 + n * 2 + (k & 1)] = v;
    }
    if (tid < NPAD) sB[tid] = (tid < SIZE_Y) ? b[tid] : 0.0f;
    __syncthreads();

    const int wave   = tid >> 5;
    const int lane   = tid & 31;
    const int lane15 = lane & 15;
    const int half   = lane >> 4;   // A: K pair select; C: row half select

    const int rowBase = blockIdx.x * 128 + wave * 16;
    const float* aPtr = x + (size_t)(rowBase + lane15) * INSIZE + half * 2;

    __builtin_prefetch(aPtr, 0, 0);

    v8f acc0 = {}; v8f acc1 = {}; v8f acc2 = {};

#pragma unroll 4
    for (int k = 0; k < INSIZE; k += 4) {
        // A fragment (16x4 f32): lane15 = M row; half selects K pair
        v2f a = *reinterpret_cast<const v2f*>(aPtr + k);
        // B fragments (4x16 f32): lane15 = N col; half selects K pair
        const float* bbase = &sW[((k >> 1) + half) * 98 + lane15 * 2];
        v2f b0 = *reinterpret_cast<const v2f*>(bbase);
        v2f b1 = *reinterpret_cast<const v2f*>(bbase + 32);
        v2f b2 = *reinterpret_cast<const v2f*>(bbase + 64);
        acc0 = __builtin_amdgcn_wmma_f32_16x16x4_f32(false, a, false, b0, (short)0, acc0, false, false);
        acc1 = __builtin_amdgcn_wmma_f32_16x16x4_f32(false, a, false, b1, (short)0, acc1, false, false);
        acc2 = __builtin_amdgcn_wmma_f32_16x16x4_f32(false, a, false, b2, (short)0, acc2, false, false);
    }

    // Bias add (column = lane15 within each N-tile)
    const float bc0 = sB[lane15];
    const float bc1 = sB[16 + lane15];
    const float bc2 = sB[32 + lane15];
#pragma unroll
    for (int v = 0; v < 8; ++v) { acc0[v] += bc0; acc1[v] += bc1; acc2[v] += bc2; }

    // Store raw y into out cols 0..39; stash cols 40..42 into LDS.
    // C layout: VGPR v holds row (half*8 + v), column lane15 of the tile.
#pragma unroll
    for (int v = 0; v < 8; ++v) {
        size_t ro = (size_t)(rowBase + half * 8 + v) * OUTC;
        out[ro + lane15]      = acc0[v];
        out[ro + 16 + lane15] = acc1[v];
        if (lane15 < 8)                 out[ro + 32 + lane15] = acc2[v];
        if (lane15 >= 8 && lane15 < 11) sMod[wave][half * 8 + v][lane15 - 8] = acc2[v];
    }
    __syncthreads();

    // Per-row mod log-softmaxes: groups {40,41}, {40,42}, {40}, {40}
    if (lane < 16) {
        const int m = lane;
        float y40 = sMod[wave][m][0];
        float y41 = sMod[wave][m][1];
        float y42 = sMod[wave][m][2];
        float m1 = fmaxf(y40, y41);
        float l1 = m1 + __logf(__expf(y40 - m1) + __expf(y41 - m1));
        float m2 = fmaxf(y40, y42);
        float l2 = m2 + __logf(__expf(y40 - m2) + __expf(y42 - m2));
        size_t ro = (size_t)(rowBase + m) * OUTC;
        out[ro + 40] = y40 - l1;
        out[ro + 41] = y41 - l1;
        out[ro + 42] = y40 - l2;
        out[ro + 43] = y42 - l2;
        out[ro + 44] = 0.0f;   // log-softmax of singleton
        out[ro + 45] = 0.0f;
    }
}

// ---------------------------------------------------------------------------
// One flip-flop step applied to a log-space state vector, with stable
// per-step normalization. Returns the normalization factor (fac).
// ---------------------------------------------------------------------------
__device__ __forceinline__ float flipflop_step(float fwd[8], const float s[40])
{
    float nw[8];
    // to-flip dests: logsumexp over all 8 sources
#pragma unroll
    for (int d = 0; d < 4; ++d) {
        float vals[8];
        float mx = -3.4e38f;
#pragma unroll
        for (int j = 0; j < 8; ++j) {
            vals[j] = fwd[j] + s[d * 8 + j];
            mx = fmaxf(mx, vals[j]);
        }
        float sum = 0.0f;
#pragma unroll
        for (int j = 0; j < 8; ++j) sum += __expf(vals[j] - mx);
        nw[d] = mx + __logf(sum);
    }
    // to-flop dests: logaddexp of matching flip/flop source
#pragma unroll
    for (int j = 0; j < 4; ++j) {
        float a = fwd[j]     + s[32 + j];
        float c = fwd[j + 4] + s[36 + j];
        float mx = fmaxf(a, c);
        nw[4 + j] = mx + __logf(__expf(a - mx) + __expf(c - mx));
    }
    float mx = nw[0];
#pragma unroll
    for (int j = 1; j < 8; ++j) mx = fmaxf(mx, nw[j]);
    float sum = 0.0f;
#pragma unroll
    for (int j = 0; j < 8; ++j) sum += __expf(nw[j] - mx);
    float fac = mx + __logf(sum);
#pragma unroll
    for (int j = 0; j < 8; ++j) fwd[j] = nw[j] - fac;
    return fac;
}

// ---------------------------------------------------------------------------
// Kernel 2a: per-(n, segment, column) transfer-matrix column.
// Pushes basis vector e_col through SEGLEN steps with per-step normalization;
// stores the normalized column (8 floats) + accumulated offset into d_ws.
// Grid: 128*SEG*8 threads = 200 blocks x 256.
// ---------------------------------------------------------------------------
__global__ __launch_bounds__(256) void scan_segments_kernel(
    const float* __restrict__ y, float* __restrict__ wsM, float* __restrict__ wsOff)
{
    const int gtid = blockIdx.x * 256 + threadIdx.x;   // 0..51199
    const int col  = gtid & 7;
    const int unit = gtid >> 3;          // 0..(NB*SEG-1)
    const int n    = unit & (NB - 1);
    const int seg  = unit >> 7;          // 0..SEG-1

    const float NEG = -1.0e30f;          // acts as -inf without NaN risk
    float fwd[8];
#pragma unroll
    for (int j = 0; j < 8; ++j) fwd[j] = (j == col) ? 0.0f : NEG;
    float offs = 0.0f;

    const int t0 = seg * SEGLEN;
    for (int tt = 0; tt < SEGLEN; ++tt) {
        const float* sc = y + ((size_t)(t0 + tt) * NB + n) * OUTC;
        float s[40];
#pragma unroll
        for (int i = 0; i < 20; ++i) {
            v2f v = *reinterpret_cast<const v2f*>(sc + 2 * i);
            s[2 * i] = v[0]; s[2 * i + 1] = v[1];
        }
        offs += flipflop_step(fwd, s);
    }

    const size_t base = (size_t)(seg * NB + n) * 8 + col;
#pragma unroll
    for (int d = 0; d < 8; ++d) wsM[base * 8 + d] = fwd[d];
    wsOff[base] = offs;
}

// ---------------------------------------------------------------------------
// Kernel 2b: per-n sequential combine of SEG segment matrices (log-matvec).
// Exactly reproduces the reference's telescoping normalization for logZ.
// ---------------------------------------------------------------------------
__global__ void scan_combine_kernel(
    const float* __restrict__ wsM, const float* __restrict__ wsOff,
    float* __restrict__ lz)
{
    const int n = threadIdx.x;            // 0..127
    const float LOG4 = 1.3862943611198906f;

    float fwd[8];
#pragma unroll
    for (int s = 0; s < 4; ++s) fwd[s] = -LOG4;
#pragma unroll
    for (int s = 4; s < 8; ++s) fwd[s] = -50000.0f - LOG4;
    float logZ = LOG4;

    for (int seg = 0; seg < SEG; ++seg) {
        const size_t ub = (size_t)(seg * NB + n) * 8;
        float g[8];
#pragma unroll
        for (int c = 0; c < 8; ++c) g[c] = fwd[c] + wsOff[ub + c];

        float nw[8];
#pragma unroll
        for (int d = 0; d < 8; ++d) {
            float vals[8];
            float mx = -3.4e38f;
#pragma unroll
            for (int c = 0; c < 8; ++c) {
                vals[c] = g[c] + wsM[(ub + c) * 8 + d];
                mx = fmaxf(mx, vals[c]);
            }
            float sum = 0.0f;
#pragma unroll
            for (int c = 0; c < 8; ++c) sum += __expf(vals[c] - mx);
            nw[d] = mx + __logf(sum);
        }
        float mx = nw[0];
#pragma unroll
        for (int d = 1; d < 8; ++d) mx = fmaxf(mx, nw[d]);
        float sum = 0.0f;
#pragma unroll
        for (int d = 0; d < 8; ++d) sum += __expf(nw[d] - mx);
        float fac = mx + __logf(sum);
#pragma unroll
        for (int d = 0; d < 8; ++d) fwd[d] = nw[d] - fac;
        logZ += fac;
    }
    lz[n] = logZ / (float)TT;
}

// ---------------------------------------------------------------------------
// Kernel 3: out[t,n,0:40] -= logZ[n]/T  (8-byte-aligned float2 RMW)
// ---------------------------------------------------------------------------
__global__ __launch_bounds__(256) void subtract_logz_kernel(
    float* __restrict__ out, const float* __restrict__ lz)
{
    int tn = blockIdx.x * 256 + threadIdx.x;
    if (tn >= NROWS) return;
    float l = lz[tn & (NB - 1)];
    float* p = out + (size_t)tn * OUTC;
#pragma unroll
    for (int i = 0; i < 20; ++i) {
        v2f v = *reinterpret_cast<v2f*>(p + 2 * i);
        v[0] -= l; v[1] -= l;
        *reinterpret_cast<v2f*>(p + 2 * i) = v;
    }
}

extern "C" void kernel_launch(void* const* d_in, const int* in_sizes, int n_in,
                              void* d_out, int out_size, void* d_ws, size_t ws_size,
                              hipStream_t stream) {
    const float* x = (const float*)d_in[0];   // (2000, 128, 256) f32
    const float* W = (const float*)d_in[1];   // (43, 256) f32
    const float* b = (const float*)d_in[2];   // (43,) f32
    float* out = (float*)d_out;               // (2000, 128, 46) f32

    // Workspace layout: lz[128] | wsM[SEG*NB*8*8] | wsOff[SEG*NB*8]  (~1.8 MB)
    float* lz    = (float*)d_ws;
    float* wsM   = lz + NB;
    float* wsOff = wsM + (size_t)SEG * NB * 64;

    gemm_wmma_kernel<<<NROWS / 128, 256, 0, stream>>>(x, W, b, out);
    scan_segments_kernel<<<(NB * SEG * 8) / 256, 256, 0, stream>>>(out, wsM, wsOff);
    scan_combine_kernel<<<1, NB, 0, stream>>>(wsM, wsOff, lz);
    subtract_logz_kernel<<<(NROWS + 255) / 256, 256, 0, stream>>>(out, lz);
}